// DGCNN_46007689674895
// MI455X (gfx1250) — compile-verified
//
#include <hip/hip_runtime.h>
#include <hip/hip_bf16.h>

// ---------------- problem constants ----------------
#define N_ATTR   100000
#define ATTR_DIM 64
#define HID      32
#define KTOP     30
#define BGR      1024
#define NPG      256
#define NNODE    (BGR * NPG)     // 262144
#define EPG      4096
#define ETOT     (BGR * EPG)     // 4194304
#define D_LAT    97              // 3*32 + 1
#define DENSE_DIM 352            // 32 * 11

// ---------------- WMMA fp32 16x16x4 ----------------
typedef float v2f __attribute__((ext_vector_type(2)));
typedef float v8f __attribute__((ext_vector_type(8)));

__device__ __forceinline__ v8f wmma4(v2f a, v2f b, v8f c) {
  // 8-arg pattern: (neg_a, A, neg_b, B, c_mod, C, reuse_a, reuse_b)
  return __builtin_amdgcn_wmma_f32_16x16x4_f32(false, a, false, b, (short)0, c,
                                               false, false);
}

// =====================================================================
// Kernel A: per-graph degrees -> ns/nd, and ew = edata[edge_id]
// =====================================================================
__global__ void __launch_bounds__(256)
k_prep(const int* __restrict__ src, const int* __restrict__ dst,
       const int* __restrict__ edge_id, const float* __restrict__ edata,
       float* __restrict__ ew, float* __restrict__ nsv, float* __restrict__ ndv) {
  __shared__ int s_do[NPG];
  __shared__ int s_di[NPG];
  const int g = blockIdx.x, tid = threadIdx.x;
  const int nbase = g * NPG, ebase = g * EPG;
  s_do[tid] = 0;
  s_di[tid] = 0;
  __syncthreads();
  for (int it = 0; it < EPG / 256; ++it) {
    int e = ebase + it * 256 + tid;
    atomicAdd(&s_do[src[e] - nbase], 1);
    atomicAdd(&s_di[dst[e] - nbase], 1);
    ew[e] = edata[edge_id[e]];   // edata is (E,1)
  }
  __syncthreads();
  float o = (float)s_do[tid]; if (o < 1.f) o = 1.f;
  float i = (float)s_di[tid]; if (i < 1.f) i = 1.f;
  nsv[nbase + tid] = 1.f / sqrtf(o);
  ndv[nbase + tid] = 1.f / sqrtf(i);
}

// =====================================================================
// Kernel B: p0 = (gathered x * ns) @ W0   (N x 128 @ 128 x 32, fp32 WMMA)
// block = 256 thr = 8 waves, handles 128 nodes (16 rows per wave)
// =====================================================================
#define XPITCH 130   // pad to avoid LDS bank conflicts on A-fragment reads

__global__ void __launch_bounds__(256)
k_xform1(const int* __restrict__ node_id, const int* __restrict__ feats,
         const float* __restrict__ ndata, const float* __restrict__ node_emb,
         const float* __restrict__ nsv, const float* __restrict__ W0,
         float* __restrict__ p) {
  extern __shared__ float smem[];
  float* s_x = smem;                    // 128 * XPITCH
  float* s_w = smem + 128 * XPITCH;     // 128 * 32
  const int tid = threadIdx.x, lane = tid & 31, wid = tid >> 5;
  const int i0 = blockIdx.x * 128;

  for (int k = tid; k < 128 * 32; k += 256) s_w[k] = W0[k];

  // stage x rows (scaled by ns): [ndata(64) | emb[feats](32) | emb[N_ATTR+nid](32)]
  for (int r = wid * 16; r < wid * 16 + 16; ++r) {
    int gi = i0 + r;
    int nid = node_id[gi];
    int ft  = feats[gi];
    float sc = nsv[gi];
    s_x[r * XPITCH + lane]       = ndata[(size_t)nid * 64 + lane] * sc;
    s_x[r * XPITCH + 32 + lane]  = ndata[(size_t)nid * 64 + 32 + lane] * sc;
    s_x[r * XPITCH + 64 + lane]  = node_emb[(size_t)ft * 32 + lane] * sc;
    s_x[r * XPITCH + 96 + lane]  = node_emb[(size_t)(N_ATTR + nid) * 32 + lane] * sc;
  }
  __syncthreads();

  // WMMA: each wave owns one 16-row M-tile; two 16-col N-tiles; K=128 in steps of 4
  const int rrow = wid * 16 + (lane & 15);
  const int n  = lane & 15;
  const int kb = (lane >> 4) * 2;   // lanes 16-31 carry K+2,K+3
  const int mh = (lane >> 4) * 8;   // C layout: lanes 16-31 hold rows M+8
  v8f acc0 = {}; v8f acc1 = {};
#pragma unroll
  for (int ks = 0; ks < 32; ++ks) {
    int k0 = ks * 4 + kb;
    v2f a;  a.x  = s_x[rrow * XPITCH + k0];  a.y  = s_x[rrow * XPITCH + k0 + 1];
    v2f b0; b0.x = s_w[k0 * 32 + n];         b0.y = s_w[(k0 + 1) * 32 + n];
    v2f b1; b1.x = s_w[k0 * 32 + 16 + n];    b1.y = s_w[(k0 + 1) * 32 + 16 + n];
    acc0 = wmma4(a, b0, acc0);
    acc1 = wmma4(a, b1, acc1);
  }
#pragma unroll
  for (int r = 0; r < 8; ++r) {
    size_t m = (size_t)(i0 + wid * 16 + mh + r);
    p[m * 32 + n]      = acc0[r];
    p[m * 32 + 16 + n] = acc1[r];
  }
}

// in-LDS 256x32 @ 32x32 transform: s_p = (s_hT^T * ns) @ s_W  (fp32 WMMA)
__device__ __forceinline__ void xform32(const float* __restrict__ s_hT,
                                        const float* __restrict__ s_W,
                                        const float* __restrict__ s_ns,
                                        float* __restrict__ s_p, int tid) {
  const int lane = tid & 31, wid = tid >> 5;
  const int rb = wid * 32;
  const int n  = lane & 15;
  const int kb = (lane >> 4) * 2;
  const int mh = (lane >> 4) * 8;
#pragma unroll
  for (int mt = 0; mt < 2; ++mt) {
    int row = rb + mt * 16 + n;
    float sc = s_ns[row];
    v8f acc0 = {}; v8f acc1 = {};
#pragma unroll
    for (int ks = 0; ks < 8; ++ks) {
      int k0 = ks * 4 + kb;
      v2f a;  a.x  = s_hT[k0 * 256 + row] * sc;
              a.y  = s_hT[(k0 + 1) * 256 + row] * sc;
      v2f b0; b0.x = s_W[k0 * 32 + n];      b0.y = s_W[(k0 + 1) * 32 + n];
      v2f b1; b1.x = s_W[k0 * 32 + 16 + n]; b1.y = s_W[(k0 + 1) * 32 + 16 + n];
      acc0 = wmma4(a, b0, acc0);
      acc1 = wmma4(a, b1, acc1);
    }
#pragma unroll
    for (int r = 0; r < 8; ++r) {
      int m = rb + mt * 16 + mh + r;
      s_p[m * 32 + n]      = acc0[r];
      s_p[m * 32 + 16 + n] = acc1[r];
    }
  }
}

// =====================================================================
// Kernel C: per-graph gconv chain (edge passes in LDS, WMMA transforms)
// dynamic LDS: p(8192) agg(8192) hT(8192) ns(256) nd(256) W(1024) b(32)
//              p4(256) a4(256)  = 26656 floats = 106,624 B
// =====================================================================
__global__ void __launch_bounds__(256)
k_graph(const int* __restrict__ src, const int* __restrict__ dst,
        const float* __restrict__ ew, const float* __restrict__ nsg,
        const float* __restrict__ ndg, const float* __restrict__ p0,
        const float* __restrict__ b0, const float* __restrict__ W1,
        const float* __restrict__ b1, const float* __restrict__ W2,
        const float* __restrict__ b2, const float* __restrict__ W3,
        const float* __restrict__ b3, float* __restrict__ feat) {
  extern __shared__ float smem[];
  float* s_p  = smem;               // [node][32] row-major (conflict-free edge reads)
  float* s_ag = smem + 8192;        // [node][32]
  float* s_hT = smem + 16384;       // [j][node] transposed (conflict-free WMMA A reads)
  float* s_ns = smem + 24576;
  float* s_nd = s_ns + 256;
  float* s_W  = s_nd + 256;
  float* s_b  = s_W + 1024;
  float* s_p4 = s_b + 32;
  float* s_a4 = s_p4 + 256;

  const int g = blockIdx.x, tid = threadIdx.x;
  const int lane = tid & 31, wid = tid >> 5;
  const int nbase = g * NPG, ebase = g * EPG;

  s_ns[tid] = nsg[nbase + tid];
  s_nd[tid] = ndg[nbase + tid];
  for (int k = tid; k < 8192; k += 256) {
    s_p[k]  = p0[(size_t)nbase * 32 + k];
    s_ag[k] = 0.f;
  }
  if (tid < 32) s_b[tid] = b0[tid];
  __syncthreads();

  for (int layer = 0; layer < 3; ++layer) {
    // edge pass: one edge per wave-step, lane j handles column j (32 banks, no conflict)
    for (int it = 0; it < EPG / 8; ++it) {
      int e = ebase + wid * (EPG / 8) + it;
      int s = src[e] - nbase;
      int d = dst[e] - nbase;
      float w = ew[e];
      atomicAdd(&s_ag[d * 32 + lane], w * s_p[s * 32 + lane]);   // ds_add_f32
    }
    __syncthreads();
    // h = tanh(agg * nd + b); write feat columns + transposed LDS copy
    {
      float ndvv = s_nd[tid];
      size_t fb = (size_t)(nbase + tid) * D_LAT + layer * 32;
#pragma unroll
      for (int j = 0; j < 32; ++j) {
        float h = tanhf(s_ag[tid * 32 + j] * ndvv + s_b[j]);
        feat[fb + j] = h;
        s_hT[j * 256 + tid] = h;
      }
    }
    __syncthreads();
    if (layer < 2) {
      const float* Wn = (layer == 0) ? W1 : W2;
      const float* bn = (layer == 0) ? b1 : b2;
      for (int k = tid; k < 1024; k += 256) s_W[k] = Wn[k];
      if (tid < 32) s_b[tid] = bn[tid];
      for (int k = tid; k < 8192; k += 256) s_ag[k] = 0.f;
      __syncthreads();
      xform32(s_hT, s_W, s_ns, s_p, tid);   // next-layer messages via WMMA
      __syncthreads();
    }
  }

  // layer 4: scalar messages p4 = (h3*ns) @ W3
  if (tid < 32) s_W[tid] = W3[tid];
  __syncthreads();
  {
    float acc = 0.f;
#pragma unroll
    for (int j = 0; j < 32; ++j) acc += s_hT[j * 256 + tid] * s_W[j];
    s_p4[tid] = acc * s_ns[tid];
    s_a4[tid] = 0.f;
  }
  __syncthreads();
  for (int it = 0; it < EPG / 256; ++it) {
    int e = ebase + it * 256 + tid;
    atomicAdd(&s_a4[dst[e] - nbase], ew[e] * s_p4[src[e] - nbase]);
  }
  __syncthreads();
  feat[(size_t)(nbase + tid) * D_LAT + 96] = tanhf(s_a4[tid] * s_nd[tid] + b3[0]);
}

// =====================================================================
// Kernel D: per-node register bitonic sort (pad to 128 with +inf),
// stable top-30 by rank, pooled rows in LDS, then the CNN/MLP head.
// =====================================================================
template <int K, int J>
__device__ __forceinline__ void ce_pass(float* v) {
#pragma unroll
  for (int i = 0; i < 128; ++i) {
    int l = i ^ J;
    if (l > i) {
      float a = v[i], b = v[l];
      bool up = (i & K) == 0;
      if (up ? (a > b) : (a < b)) { v[i] = b; v[l] = a; }
    }
  }
}
template <int K, int J>
__device__ __forceinline__ void jloop(float* v) {
  ce_pass<K, J>(v);
  if constexpr (J > 1) jloop<K, J / 2>(v);
}
template <int K>
__device__ __forceinline__ void kloop(float* v) {
  jloop<K, K / 2>(v);
  if constexpr (K < 128) kloop<K * 2>(v);
}

__global__ void __launch_bounds__(256)
k_head(const float* __restrict__ feat,
       const float* __restrict__ c1w, const float* __restrict__ c1b,
       const float* __restrict__ c2w, const float* __restrict__ c2b,
       const float* __restrict__ f1w, const float* __restrict__ f1b,
       const float* __restrict__ f2w, const float* __restrict__ f2b,
       float* __restrict__ out) {
  __shared__ float s_keys[NPG];
  __shared__ float s_z[KTOP * D_LAT];   // 2910: pooled rows = z for conv1
  __shared__ float s_a[16 * 30];
  __shared__ float s_m[16 * 15];
  __shared__ float s_c[DENSE_DIM];
  __shared__ float s_f[128];

  const int g = blockIdx.x, tid = threadIdx.x;
  const size_t node = (size_t)g * NPG + tid;

  float v[128];
#pragma unroll
  for (int j = 0; j < D_LAT; ++j) v[j] = feat[node * D_LAT + j];
#pragma unroll
  for (int j = D_LAT; j < 128; ++j) v[j] = __builtin_inff();
  kloop<2>(v);                       // ascending; +inf padding sinks to the end
  float key = v[96];                 // row max = sorted[-1]
  s_keys[tid] = key;
  __syncthreads();

  // stable descending rank (matches lax.top_k tie-breaking by lower index)
  int rank = 0;
  for (int j = 0; j < NPG; ++j) {
    float kj = s_keys[j];
    rank += (kj > key) || (kj == key && j < tid);
  }
  if (rank < KTOP) {
#pragma unroll
    for (int j = 0; j < D_LAT; ++j) s_z[rank * D_LAT + j] = v[j];
  }
  __syncthreads();

  // conv1 (16,1,97) stride 97: out1[f][t] = relu(dot(z_row_t, w_f) + b_f)
  for (int o = tid; o < 16 * 30; o += 256) {
    int f = o / 30, t = o % 30;
    float acc = c1b[f];
    for (int j = 0; j < D_LAT; ++j) acc += s_z[t * D_LAT + j] * c1w[f * D_LAT + j];
    s_a[f * 30 + t] = fmaxf(acc, 0.f);
  }
  __syncthreads();
  // maxpool /2
  for (int o = tid; o < 16 * 15; o += 256) {
    int f = o / 15, u = o % 15;
    s_m[o] = fmaxf(s_a[f * 30 + 2 * u], s_a[f * 30 + 2 * u + 1]);
  }
  __syncthreads();
  // conv2 (32,16,5) VALID: 11 positions
  for (int o = tid; o < DENSE_DIM; o += 256) {
    int f2 = o / 11, vv = o % 11;
    float acc = c2b[f2];
    for (int f = 0; f < 16; ++f)
#pragma unroll
      for (int dt = 0; dt < 5; ++dt)
        acc += s_m[f * 15 + vv + dt] * c2w[f2 * 80 + f * 5 + dt];
    s_c[o] = fmaxf(acc, 0.f);
  }
  __syncthreads();
  // fc1 (128,352) + relu
  if (tid < 128) {
    float acc = f1b[tid];
    for (int j = 0; j < DENSE_DIM; ++j) acc += s_c[j] * f1w[tid * DENSE_DIM + j];
    s_f[tid] = fmaxf(acc, 0.f);
  }
  __syncthreads();
  // fc2 (1,128)
  if (tid == 0) {
    float acc = f2b[0];
    for (int j = 0; j < 128; ++j) acc += s_f[j] * f2w[j];
    out[g] = acc;
  }
}

// =====================================================================
// launch
// =====================================================================
extern "C" void kernel_launch(void* const* d_in, const int* in_sizes, int n_in,
                              void* d_out, int out_size, void* d_ws, size_t ws_size,
                              hipStream_t stream) {
  const int*   feats    = (const int*)d_in[0];
  const int*   node_id  = (const int*)d_in[1];
  const int*   edge_id  = (const int*)d_in[2];
  const int*   src      = (const int*)d_in[3];
  const int*   dst      = (const int*)d_in[4];
  const float* ndata    = (const float*)d_in[5];
  const float* edata    = (const float*)d_in[6];
  const float* node_emb = (const float*)d_in[7];
  const float* W0 = (const float*)d_in[8];
  const float* b0 = (const float*)d_in[9];
  const float* W1 = (const float*)d_in[10];
  const float* b1 = (const float*)d_in[11];
  const float* W2 = (const float*)d_in[12];
  const float* b2 = (const float*)d_in[13];
  const float* W3 = (const float*)d_in[14];
  const float* b3 = (const float*)d_in[15];
  const float* c1w = (const float*)d_in[16];
  const float* c1b = (const float*)d_in[17];
  const float* c2w = (const float*)d_in[18];
  const float* c2b = (const float*)d_in[19];
  const float* f1w = (const float*)d_in[20];
  const float* f1b = (const float*)d_in[21];
  const float* f2w = (const float*)d_in[22];
  const float* f2b = (const float*)d_in[23];

  // workspace layout (floats): ew[E] ns[N] nd[N] p[N*32] feat[N*97] ~154 MB
  float* ws   = (float*)d_ws;
  float* ew   = ws;
  float* nsv  = ew + (size_t)ETOT;
  float* ndv  = nsv + (size_t)NNODE;
  float* p    = ndv + (size_t)NNODE;
  float* feat = p + (size_t)NNODE * 32;

  k_prep<<<BGR, 256, 0, stream>>>(src, dst, edge_id, edata, ew, nsv, ndv);

  size_t smemB = (size_t)(128 * XPITCH + 128 * 32) * sizeof(float);   // ~83 KB
  k_xform1<<<NNODE / 128, 256, smemB, stream>>>(node_id, feats, ndata, node_emb,
                                                nsv, W0, p);

  size_t smemC = (size_t)(8192 * 3 + 256 * 2 + 1024 + 32 + 256 * 2) * sizeof(float); // ~104 KB
  k_graph<<<BGR, 256, smemC, stream>>>(src, dst, ew, nsv, ndv, p,
                                       b0, W1, b1, W2, b2, W3, b3, feat);

  k_head<<<BGR, 256, 0, stream>>>(feat, c1w, c1b, c2w, c2b, f1w, f1b, f2w, f2b,
                                  (float*)d_out);
}